// CenterLoss_32701880992422
// MI455X (gfx1250) — compile-verified
//
#include <hip/hip_runtime.h>

// Problem constants (match reference)
#define BATCH      8192
#define NUM_CLS    751
#define FEAT_DIM   2048
#define CLIP_LO    1e-12f
#define CLIP_HI    1e12f

typedef float v2f __attribute__((ext_vector_type(2)));
typedef float v8f __attribute__((ext_vector_type(8)));

// Wave32 all-lanes reduction using V_WMMA_F32_16X16X4_F32 with an all-ones
// B matrix. A layout (16x4 f32): lane l<16 -> row l holds K0/K1 in v[0]/v[1];
// lane l+16 -> row l holds K2/K3. With a = {acc, 0}, D[m][n] = acc[m]+acc[m+16].
// C/D layout: VGPR r, lanes 0-15 = M=r, lanes 16-31 = M=r+8. Summing the 8
// result VGPRs gives the lower/upper half-sum per half-wave; one xor-16
// shuffle completes the reduction and broadcasts it to all 32 lanes.
// Requires EXEC all-ones at the call site (all call sites are convergent).
__device__ __forceinline__ float wave_reduce_sum(float v) {
    v2f a; a[0] = v;    a[1] = 0.0f;
    v2f b; b[0] = 1.0f; b[1] = 1.0f;
    v8f c = {};
    c = __builtin_amdgcn_wmma_f32_16x16x4_f32(
            /*neg_a=*/false, a, /*neg_b=*/false, b,
            /*c_mod=*/(short)0, c, /*reuse_a=*/false, /*reuse_b=*/false);
    float s = c[0] + c[1] + c[2] + c[3] + c[4] + c[5] + c[6] + c[7];
    s += __shfl_xor(s, 16);
    return s;
}

// One wave (32 lanes) per batch row. 8 waves per block.
__global__ __launch_bounds__(256)
void center_loss_rows(const float* __restrict__ x,
                      const float* __restrict__ labels,
                      const float* __restrict__ centers,
                      float* __restrict__ row_out) {
    const int lane = threadIdx.x & 31;
    const int wave = threadIdx.x >> 5;
    const int row  = blockIdx.x * 8 + wave;

    // ---- Phase A: recover class index from the one-hot row (exact) ----
    const float* lr = labels + (size_t)row * NUM_CLS;
    float idx_part = 0.0f;
    for (int k = lane; k < NUM_CLS; k += 32) {
        float v = lr[k];
        if (v > 0.5f) idx_part = (float)k;   // exactly one lane sees the 1.0
    }
    const int cidx = (int)wave_reduce_sum(idx_part);  // exact: 0..750 < 2^24

    // ---- Phase B: sum((x - centers[cidx])^2) over 2048 features ----
    const float4* xr = (const float4*)(x + (size_t)row  * FEAT_DIM);
    const float4* cr = (const float4*)(centers + (size_t)cidx * FEAT_DIM);
    float acc = 0.0f;
    #pragma unroll
    for (int i = 0; i < FEAT_DIM / (4 * 32); ++i) {   // 16 iterations
        const int e = lane + 32 * i;
        float4 xv = xr[e];
        float4 cv = cr[e];
        float d0 = xv.x - cv.x;
        float d1 = xv.y - cv.y;
        float d2 = xv.z - cv.z;
        float d3 = xv.w - cv.w;
        acc = fmaf(d0, d0, acc);
        acc = fmaf(d1, d1, acc);
        acc = fmaf(d2, d2, acc);
        acc = fmaf(d3, d3, acc);
    }

    float s = wave_reduce_sum(acc);
    s = fminf(fmaxf(s, CLIP_LO), CLIP_HI);
    if (lane == 0) row_out[row] = s;
}

// Deterministic final reduction: 8192 per-row values -> mean (single block).
__global__ __launch_bounds__(256)
void reduce_mean(const float* __restrict__ row_vals, float* __restrict__ out) {
    __shared__ float wsum[8];
    const int t    = threadIdx.x;
    const int lane = t & 31;
    const int wave = t >> 5;

    float acc = 0.0f;
    for (int i = t; i < BATCH; i += 256) acc += row_vals[i];

    float s = wave_reduce_sum(acc);
    if (lane == 0) wsum[wave] = s;
    __syncthreads();

    if (t == 0) {
        float tot = 0.0f;
        #pragma unroll
        for (int w = 0; w < 8; ++w) tot += wsum[w];
        out[0] = tot * (1.0f / (float)BATCH);
    }
}

extern "C" void kernel_launch(void* const* d_in, const int* in_sizes, int n_in,
                              void* d_out, int out_size, void* d_ws, size_t ws_size,
                              hipStream_t stream) {
    const float* x       = (const float*)d_in[0];   // [8192, 2048] f32
    const float* labels  = (const float*)d_in[1];   // [8192, 751]  f32 one-hot
    const float* centers = (const float*)d_in[2];   // [751, 2048]  f32
    float* out = (float*)d_out;                     // scalar f32
    float* ws  = (float*)d_ws;                      // 8192 f32 = 32 KB scratch

    center_loss_rows<<<BATCH / 8, 256, 0, stream>>>(x, labels, centers, ws);
    reduce_mean<<<1, 256, 0, stream>>>(ws, out);
}